// proj_CommunityArticleGNNEncoder_59785944760472
// MI455X (gfx1250) — compile-verified
//
#include <hip/hip_runtime.h>

typedef __attribute__((ext_vector_type(2))) float v2f;
typedef __attribute__((ext_vector_type(8))) float v8f;

// ---------------------------------------------------------------------------
// Generic fp32 WMMA GEMM:  D = act( A[M,K] @ B[K,N] + epilogue )
// Templated on S = number of 16-row subtiles per wave (branch-free K-loop).
// Block = 256 threads = 8 waves; block tile = (16*S) rows x 128 cols.
// Each wave: S row-subtiles x 16 cols; K-loop steps of 4 with
// v_wmma_f32_16x16x4_f32, B fragment reused across the S subtiles.
// Integer offsets (not pointer arrays) keep loads on the GLOBAL path.
// Caller guarantees all S subtiles are in-range (tail handled by S=1 launch).
// All branches are wave-uniform so EXEC is all-ones at every WMMA.
// ---------------------------------------------------------------------------
template <int S>
__global__ __launch_bounds__(256)
void gemm_wmma_f32(const float* __restrict__ A, const float* __restrict__ B,
                   float* __restrict__ D,
                   const float* __restrict__ bias,   // [N] always added
                   const float* __restrict__ gbias,  // [N] added iff cnt>0
                   const float* __restrict__ addm,   // [M,N] addend
                   const float* __restrict__ cnt,    // [M] divides addm
                   int row_base, int N, int K, int relu)
{
    const int lane = threadIdx.x & 31;
    const int wave = threadIdx.x >> 5;
    const int half = lane >> 4;     // 0: lanes 0-15, 1: lanes 16-31
    const int l16  = lane & 15;
    const int row0 = row_base + blockIdx.y * (16 * S);
    const int col0 = blockIdx.x * 128 + wave * 16;
    if (col0 >= N) return;          // wave-uniform

    v8f acc[S];
#pragma unroll
    for (int s = 0; s < S; ++s) acc[s] = v8f{};

    // A fragment (16x4, 32-bit layout): lane holds A[row+l16][ka], [ka+1]
    // with ka = k0 + 2*half.  B fragment (4x16): VGPR0=row ka, VGPR1=row ka+1.
    size_t aoff[S];
#pragma unroll
    for (int s = 0; s < S; ++s)
        aoff[s] = (size_t)(row0 + s * 16 + l16) * K + (size_t)(half * 2);
    const size_t boff = (size_t)(half * 2) * N + (size_t)(col0 + l16);

#pragma unroll 4
    for (int k0 = 0; k0 < K; k0 += 4) {
        v2f b;
        b.x = B[boff + (size_t)k0 * N];
        b.y = B[boff + (size_t)(k0 + 1) * N];
        v2f a[S];
#pragma unroll
        for (int s = 0; s < S; ++s)
            a[s] = *(const v2f*)(A + aoff[s] + k0);
#pragma unroll
        for (int s = 0; s < S; ++s)
            acc[s] = __builtin_amdgcn_wmma_f32_16x16x4_f32(
                false, a[s], false, b, (short)0, acc[s], false, false);
    }

    const int n = col0 + l16;
#pragma unroll
    for (int s = 0; s < S; ++s) {
        const int rbase = row0 + s * 16;
#pragma unroll
        for (int i = 0; i < 8; ++i) {
            // C/D layout: VGPR i -> row (i + 8*half), col l16
            const int m = rbase + i + half * 8;
            float v = acc[s][i];
            if (addm) {
                float av = addm[(size_t)m * N + n];
                if (cnt) {
                    const float c = cnt[m];
                    av = av / fmaxf(c, 1.0f);
                    if (gbias && c > 0.0f) v += gbias[n];
                }
                v += av;
            }
            if (bias) v += bias[n];
            if (relu) v = fmaxf(v, 0.0f);
            D[(size_t)m * N + n] = v;
        }
    }
}

// ---------------------------------------------------------------------------
// Segment-sum scatter: agg[dst] += X[src] (256 floats/row), cnt[dst] += 1.
// 4 edges per 256-thread block; 64 threads/edge, float4 per thread.
// ---------------------------------------------------------------------------
__global__ __launch_bounds__(256)
void scatter_add_256(const float* __restrict__ X, const int* __restrict__ ei,
                     int E, float* __restrict__ agg, float* __restrict__ cnt)
{
    const int e = blockIdx.x * 4 + (threadIdx.x >> 6);
    if (e >= E) return;
    const int src = ei[e];
    const int dst = ei[E + e];
    const int f = (threadIdx.x & 63) * 4;
    const float4 v = *(const float4*)(X + (size_t)src * 256 + f);
    float* o = agg + (size_t)dst * 256 + f;
    atomicAdd(o + 0, v.x);
    atomicAdd(o + 1, v.y);
    atomicAdd(o + 2, v.z);
    atomicAdd(o + 3, v.w);
    if ((threadIdx.x & 63) == 0) atomicAdd(cnt + dst, 1.0f);
}

// agg[m][:] /= max(cnt[m],1)  (in place), width 256
__global__ __launch_bounds__(256)
void divide_by_cnt(float* __restrict__ agg, const float* __restrict__ cnt, int M)
{
    const int idx = blockIdx.x * 256 + threadIdx.x;
    if (idx >= M * 256) return;
    agg[idx] = agg[idx] / fmaxf(cnt[idx >> 8], 1.0f);
}

// out[n] = (base ? base[n] : 0) + sum_k v[k] * W[k*N + n]
__global__ __launch_bounds__(256)
void vecproj(const float* __restrict__ v, const float* __restrict__ W,
             const float* __restrict__ base, float* __restrict__ out, int K, int N)
{
    const int n = blockIdx.x * 256 + threadIdx.x;
    if (n >= N) return;
    float s = base ? base[n] : 0.0f;
    for (int k = 0; k < K; ++k) s += v[k] * W[(size_t)k * N + n];
    out[n] = s;
}

extern "C" void kernel_launch(void* const* d_in, const int* in_sizes, int n_in,
                              void* d_out, int out_size, void* d_ws, size_t ws_size,
                              hipStream_t stream)
{
    (void)n_in; (void)out_size; (void)ws_size;
    const int M = 50000, F = 768, P = 1024, H = 256, O = 128;

    const float* article_x   = (const float*)d_in[0];
    const float* community_x = (const float*)d_in[1];
    const int*   e_wb        = (const int*)  d_in[2];
    const int*   e_mb        = (const int*)  d_in[3];
    const int*   e_int       = (const int*)  d_in[4];
    const float* W1  = (const float*)d_in[5];
    const float* b1  = (const float*)d_in[6];
    const float* W2  = (const float*)d_in[7];
    const float* b2  = (const float*)d_in[8];
    const float* Wl1 = (const float*)d_in[9];
    const float* bl1 = (const float*)d_in[10];
    const float* Wr1 = (const float*)d_in[11];
    const float* Wl2 = (const float*)d_in[12];
    const float* bl2 = (const float*)d_in[13];
    const float* Wr2 = (const float*)d_in[14];
    const float* Wl3 = (const float*)d_in[15];
    const float* bl3 = (const float*)d_in[16];
    const float* Wr3 = (const float*)d_in[17];
    const float* W3  = (const float*)d_in[18];
    const float* b3  = (const float*)d_in[19];

    const int E_wb  = in_sizes[2] / 2;
    const int E_mb  = in_sizes[3] / 2;
    const int E_int = in_sizes[4] / 2;

    // Workspace carve-up (floats): ~156 MB total
    float* ws = (float*)d_ws;
    size_t off = 0;
    float* A1  = ws + off; off += (size_t)F * H;   // W1@Wl1
    float* A2  = ws + off; off += (size_t)F * H;   // W1@Wl2
    float* B1  = ws + off; off += (size_t)F * H;   // W2@Wr1
    float* g1  = ws + off; off += H;               // b1@Wl1  (gated on cnt>0)
    float* g2  = ws + off; off += H;               // b1@Wl2  (gated on cnt>0)
    float* c1  = ws + off; off += H;               // bl1 + b2@Wr1
    float* R0  = ws + off; off += (size_t)M * H;   // p1 -> h1 -> t
    float* R1  = ws + off; off += (size_t)M * H;   // p2 -> h2 -> h3
    float* AGG = ws + off; off += (size_t)M * H;   // segment sums
    float* CNT = ws + off; off += (size_t)M;       // segment counts

    const dim3 blk(256);
    auto gemm = [&](const float* A, const float* Bm, float* D,
                    const float* bias, const float* gbias,
                    const float* addm, const float* cntp,
                    int m, int n, int k, int relu) {
        const int mb = m / 64;                 // full 64-row blocks
        if (mb > 0) {
            dim3 grid((n + 127) / 128, mb);
            gemm_wmma_f32<4><<<grid, blk, 0, stream>>>(
                A, Bm, D, bias, gbias, addm, cntp, 0, n, k, relu);
        }
        const int mr = m - mb * 64;            // tail rows (multiple of 16)
        if (mr > 0) {
            dim3 grid((n + 127) / 128, mr / 16);
            gemm_wmma_f32<1><<<grid, blk, 0, stream>>>(
                A, Bm, D, bias, gbias, addm, cntp, mb * 64, n, k, relu);
        }
    };

    // --- fold projection weights into conv weights (exact in real arithmetic)
    gemm(W1, Wl1, A1, nullptr, nullptr, nullptr, nullptr, F, H, P, 0);
    gemm(W1, Wl2, A2, nullptr, nullptr, nullptr, nullptr, F, H, P, 0);
    gemm(W2, Wr1, B1, nullptr, nullptr, nullptr, nullptr, F, H, P, 0);
    hipLaunchKernelGGL(vecproj, dim3(1), blk, 0, stream, b1, Wl1, (const float*)nullptr, g1, P, H);
    hipLaunchKernelGGL(vecproj, dim3(1), blk, 0, stream, b1, Wl2, (const float*)nullptr, g2, P, H);
    hipLaunchKernelGGL(vecproj, dim3(1), blk, 0, stream, b2, Wr1, bl1, c1, P, H);

    // --- project article features straight to HID (skips the 1024-wide stage)
    gemm(article_x, A1, R0, nullptr, nullptr, nullptr, nullptr, M, H, F, 0); // p1
    gemm(article_x, A2, R1, nullptr, nullptr, nullptr, nullptr, M, H, F, 0); // p2

    // --- conv1: h1 = relu(mean_wb(p1) + community_x@B1 + [cnt>0]g1 + c1)
    hipMemsetAsync(AGG, 0, (size_t)M * H * sizeof(float), stream);
    hipMemsetAsync(CNT, 0, (size_t)M * sizeof(float), stream);
    hipLaunchKernelGGL(scatter_add_256, dim3((E_wb + 3) / 4), blk, 0, stream,
                       R0, e_wb, E_wb, AGG, CNT);
    gemm(community_x, B1, R0, c1, g1, AGG, CNT, M, H, F, 1);                 // h1 -> R0

    // --- conv2: h2 = relu(mean_mb(p2) + h1@Wr2 + [cnt>0]g2 + bl2)
    hipMemsetAsync(AGG, 0, (size_t)M * H * sizeof(float), stream);
    hipMemsetAsync(CNT, 0, (size_t)M * sizeof(float), stream);
    hipLaunchKernelGGL(scatter_add_256, dim3((E_mb + 3) / 4), blk, 0, stream,
                       R1, e_mb, E_mb, AGG, CNT);
    gemm(R0, Wr2, R1, bl2, g2, AGG, CNT, M, H, H, 1);                        // h2 -> R1

    // --- conv3: h3 = relu(mean_int(h2)@Wl3 + community_x@Wr3 + bl3)
    hipMemsetAsync(AGG, 0, (size_t)M * H * sizeof(float), stream);
    hipMemsetAsync(CNT, 0, (size_t)M * sizeof(float), stream);
    hipLaunchKernelGGL(scatter_add_256, dim3((E_int + 3) / 4), blk, 0, stream,
                       R1, e_int, E_int, AGG, CNT);
    hipLaunchKernelGGL(divide_by_cnt, dim3((M * H + 255) / 256), blk, 0, stream,
                       AGG, CNT, M);                                         // AGG = mean3
    gemm(AGG, Wl3, R0, nullptr, nullptr, nullptr, nullptr, M, H, H, 0);      // t  -> R0
    gemm(community_x, Wr3, R1, bl3, nullptr, R0, nullptr, M, H, F, 1);       // h3 -> R1

    // --- final linear
    gemm(R1, W3, (float*)d_out, b3, nullptr, nullptr, nullptr, M, O, H, 0);
}